// ooi_net_67774583931064
// MI455X (gfx1250) — compile-verified
//
#include <hip/hip_runtime.h>

typedef float v2f __attribute__((ext_vector_type(2)));
typedef float v4f __attribute__((ext_vector_type(4)));
typedef float v8f __attribute__((ext_vector_type(8)));

#define NGRAPH 512
#define NPG    64      // nodes per graph
#define DFEAT  256
#define PPG    512     // pairs per graph
#define EPG    1024    // edges per graph (64*16)
#define HDIM   512
#define NTOT   (NGRAPH * NPG)    // 32768
#define ETOT   (NGRAPH * EPG)    // 524288
#define NPAIR  (NGRAPH * PPG)    // 262144

#define WMMA_F32(a, b, c) \
  __builtin_amdgcn_wmma_f32_16x16x4_f32(false, (a), false, (b), (short)0, (c), false, false)

// A 16x4 fragment from LDS: lanes 0-15 -> M=lane,K=k0+{0,1}; lanes 16-31 -> M=lane-16,K=k0+{2,3}
__device__ __forceinline__ v2f ld_a_frag(const float* base, int m0, int k0, int ld, int lane) {
  int m  = m0 + (lane & 15);
  int kk = k0 + ((lane >> 4) << 1);
  return *(const v2f*)(base + m * ld + kk);   // ds_load_b64
}

// ---------------------------------------------------------------------------
// Weight repack: row-major W[K x Nreal]  ->  fragment-major P
// Fragment-pair fp = n16 * (K/8) + k0/8 ; each lane's 4 dwords contiguous:
//   P[fp*128 + lane*4 + {0,1,2,3}] =
//     { W[k0+2*(l>>4)][col], W[k0+2*(l>>4)+1][col],
//       W[k0+4+2*(l>>4)][col], W[k0+4+2*(l>>4)+1][col] }   (col = n0 + (l&15))
// Zero-pads columns >= Nreal (used to pad w2's N=17 -> 32).
// ---------------------------------------------------------------------------
__global__ void repack_b_kernel(const float* __restrict__ W, float* __restrict__ P,
                                int K, int Nreal, int Npad) {
  int tid = blockIdx.x * blockDim.x + threadIdx.x;
  int numKp = K >> 3;
  int total = numKp * (Npad >> 4) * 32;
  if (tid >= total) return;
  int l  = tid & 31;
  int fp = tid >> 5;
  int kp  = fp % numKp;
  int n16 = fp / numKp;
  int k0  = kp << 3;
  int col = (n16 << 4) + (l & 15);
  int r0  = k0 + ((l >> 4) << 1);
  v4f v;
  if (col < Nreal) {
    v.x = W[(r0 + 0) * Nreal + col];
    v.y = W[(r0 + 1) * Nreal + col];
    v.z = W[(r0 + 4) * Nreal + col];
    v.w = W[(r0 + 5) * Nreal + col];
  } else {
    v = v4f{0.0f, 0.0f, 0.0f, 0.0f};
  }
  *(v4f*)(P + (size_t)fp * 128 + l * 4) = v;   // global_store_b128
}

// Packed B fetch: one global_load_b128 feeds two WMMAs
__device__ __forceinline__ v4f ld_bp(const float* __restrict__ P, int fp, int lane) {
  return *(const v4f*)(P + (size_t)fp * 128 + lane * 4);
}

// ---- degree = 1 (self loop) + in-degree ----
__global__ void deg_init_kernel(float* __restrict__ deg) {
  int i = blockIdx.x * blockDim.x + threadIdx.x;
  if (i < NTOT) deg[i] = 1.0f;
}
__global__ void deg_count_kernel(const int* __restrict__ dst, float* __restrict__ deg) {
  int i = blockIdx.x * blockDim.x + threadIdx.x;
  if (i < ETOT) atomicAdd(&deg[dst[i]], 1.0f);
}

// ---- one GCN layer: per-graph LDS accumulation + WMMA matmul ----
template <bool RELU>
__global__ __launch_bounds__(256)
void gcn_layer_kernel(const float* __restrict__ hin,
                      const int* __restrict__ src,
                      const int* __restrict__ dst,
                      const float* __restrict__ deg,
                      const float* __restrict__ Wp,   // packed [256x256], numKp=32
                      const float* __restrict__ bias,
                      float* __restrict__ hout) {
  extern __shared__ float agg[];           // NPG * DFEAT floats = 64 KB
  const int g = blockIdx.x;
  const int tid = threadIdx.x;
  const int lane = tid & 31;
  const int wave = tid >> 5;
  const int node0 = g * NPG;

  // self loop: agg[n] = h[n] / deg[n]
  for (int i = tid; i < NPG * (DFEAT / 4); i += 256) {
    int n  = i >> 6;
    int c4 = (i & 63) << 2;
    int gn = node0 + n;
    v4f xv = *(const v4f*)(hin + gn * DFEAT + c4);
    float s = 1.0f / deg[gn];
    *(v4f*)(agg + n * DFEAT + c4) = xv * s;
  }
  __syncthreads();

  // edges: one wave per edge, lanes cover features in two float4 passes
  const int ebase = g * EPG;
  for (int e = wave; e < EPG; e += 8) {
    int s = src[ebase + e];
    int d = dst[ebase + e];
    float nrm = rsqrtf(deg[s] * deg[d]);
    int dl = d - node0;
    const float* xs = hin + s * DFEAT;
#pragma unroll
    for (int half = 0; half < 2; ++half) {
      int c = half * 128 + lane * 4;
      v4f xv = *(const v4f*)(xs + c);
      atomicAdd(&agg[dl * DFEAT + c + 0], xv.x * nrm);
      atomicAdd(&agg[dl * DFEAT + c + 1], xv.y * nrm);
      atomicAdd(&agg[dl * DFEAT + c + 2], xv.z * nrm);
      atomicAdd(&agg[dl * DFEAT + c + 3], xv.w * nrm);
    }
  }
  __syncthreads();

  // matmul 64x256 = agg @ W + bias  (M=64 N=256 K=256, numKp = 32)
  const int m_tile  = wave & 3;
  const int n_wave  = (wave >> 2) * 8;     // in n16 units: waves 0-3 tiles 0-7, 4-7 tiles 8-15
  const int n_local = lane & 15;
  const int m_hi    = (lane >> 4) << 3;

  for (int grp = 0; grp < 2; ++grp) {
    int n16b = n_wave + grp * 4;
    v8f acc[4] = {v8f{0}, v8f{0}, v8f{0}, v8f{0}};
    for (int k0 = 0; k0 < DFEAT; k0 += 8) {
      v2f a0 = ld_a_frag(agg, m_tile * 16, k0, DFEAT, lane);
      v2f a1 = ld_a_frag(agg, m_tile * 16, k0 + 4, DFEAT, lane);
      int kp = k0 >> 3;
#pragma unroll
      for (int t = 0; t < 4; ++t) {
        v4f bq = ld_bp(Wp, (n16b + t) * 32 + kp, lane);
        v2f b0 = {bq.x, bq.y};
        v2f b1 = {bq.z, bq.w};
        acc[t] = WMMA_F32(a0, b0, acc[t]);
        acc[t] = WMMA_F32(a1, b1, acc[t]);
      }
    }
#pragma unroll
    for (int t = 0; t < 4; ++t) {
      int n = (n16b + t) * 16 + n_local;
      float bv = bias[n];
#pragma unroll
      for (int r = 0; r < 8; ++r) {
        float v = acc[t][r] + bv;
        if (RELU) v = fmaxf(v, 0.0f);
        hout[(node0 + m_tile * 16 + m_hi + r) * DFEAT + n] = v;
      }
    }
  }
}

// ---- fused pair gather + 2-layer MLP head ----
__global__ __launch_bounds__(256)
void pair_mlp_kernel(const float* __restrict__ emb,
                     const int* __restrict__ pairs,
                     const float* __restrict__ w1p,  // packed [256 x 512], numKp=32
                     const float* __restrict__ b1,
                     const float* __restrict__ w2p,  // packed [512 x 32(pad)], numKp=64
                     const float* __restrict__ b2,
                     float* __restrict__ out, int out_dim) {
  extern __shared__ float smem[];
  float* ci  = smem;                       // 64*256 = 64 KB
  float* hid = smem + 64 * DFEAT;          // 64*512 = 128 KB
  const int tid = threadIdx.x;
  const int lane = tid & 31;
  const int wave = tid >> 5;
  const int row0 = blockIdx.x * 64;

  // gather: ci = 0.5*(emb[a] + emb[b])   (float4 loads/stores)
  for (int i = tid; i < 64 * (DFEAT / 4); i += 256) {
    int r  = i >> 6;
    int c4 = (i & 63) << 2;
    int pid = row0 + r;
    int gph = pid >> 9;                    // / PPG
    int p   = pid & (PPG - 1);
    int pa = pairs[(gph * PPG + p) * 2 + 0] + gph * NPG;
    int pb = pairs[(gph * PPG + p) * 2 + 1] + gph * NPG;
    v4f ea = *(const v4f*)(emb + pa * DFEAT + c4);
    v4f eb = *(const v4f*)(emb + pb * DFEAT + c4);
    v4f cv = (ea + eb) * 0.5f;
    *(v4f*)(ci + r * DFEAT + c4) = cv;
  }
  __syncthreads();

  const int m_tile  = wave & 3;
  const int n_wave  = (wave >> 2) * 16;    // in n16 units: 0..15 or 16..31
  const int n_local = lane & 15;
  const int m_hi    = (lane >> 4) << 3;

  // layer 1: hidden = relu(ci @ w1 + b1)   M=64 N=512 K=256 (numKp=32)
  for (int grp = 0; grp < 4; ++grp) {
    int n16b = n_wave + grp * 4;
    v8f acc[4] = {v8f{0}, v8f{0}, v8f{0}, v8f{0}};
    for (int k0 = 0; k0 < DFEAT; k0 += 8) {
      v2f a0 = ld_a_frag(ci, m_tile * 16, k0, DFEAT, lane);
      v2f a1 = ld_a_frag(ci, m_tile * 16, k0 + 4, DFEAT, lane);
      int kp = k0 >> 3;
#pragma unroll
      for (int t = 0; t < 4; ++t) {
        v4f bq = ld_bp(w1p, (n16b + t) * 32 + kp, lane);
        v2f b0 = {bq.x, bq.y};
        v2f b1v = {bq.z, bq.w};
        acc[t] = WMMA_F32(a0, b0, acc[t]);
        acc[t] = WMMA_F32(a1, b1v, acc[t]);
      }
    }
#pragma unroll
    for (int t = 0; t < 4; ++t) {
      int n = (n16b + t) * 16 + n_local;
      float bv = b1[n];
#pragma unroll
      for (int r = 0; r < 8; ++r) {
        float v = fmaxf(acc[t][r] + bv, 0.0f);
        hid[(m_tile * 16 + m_hi + r) * HDIM + n] = v;
      }
    }
  }
  __syncthreads();

  // layer 2: out = hidden @ w2 + b2   M=64 N=out_dim (padded 32) K=512 (numKp=64)
  {
    int n16 = wave >> 2;                   // waves 0-3: n16=0, waves 4-7: n16=1
    v8f acc = v8f{0};
    for (int k0 = 0; k0 < HDIM; k0 += 8) {
      v2f a0 = ld_a_frag(hid, m_tile * 16, k0, HDIM, lane);
      v2f a1 = ld_a_frag(hid, m_tile * 16, k0 + 4, HDIM, lane);
      int kp = k0 >> 3;
      v4f bq = ld_bp(w2p, n16 * 64 + kp, lane);
      v2f b0 = {bq.x, bq.y};
      v2f b1v = {bq.z, bq.w};
      acc = WMMA_F32(a0, b0, acc);
      acc = WMMA_F32(a1, b1v, acc);
    }
    int n = n16 * 16 + n_local;
    if (n < out_dim) {
      float bv = b2[n];
#pragma unroll
      for (int r = 0; r < 8; ++r)
        out[(row0 + m_tile * 16 + m_hi + r) * out_dim + n] = acc[r] + bv;
    }
  }
}

extern "C" void kernel_launch(void* const* d_in, const int* in_sizes, int n_in,
                              void* d_out, int out_size, void* d_ws, size_t ws_size,
                              hipStream_t stream) {
  (void)in_sizes; (void)n_in; (void)out_size; (void)ws_size;
  const float* x      = (const float*)d_in[0];
  const int*   edge   = (const int*)  d_in[1];
  const int*   pairs  = (const int*)  d_in[2];
  const float* gcn_w1 = (const float*)d_in[3];
  const float* gcn_b1 = (const float*)d_in[4];
  const float* gcn_w2 = (const float*)d_in[5];
  const float* gcn_b2 = (const float*)d_in[6];
  const float* lr_w1  = (const float*)d_in[7];
  const float* lr_b1  = (const float*)d_in[8];
  const float* lr_w2  = (const float*)d_in[9];
  const float* lr_b2  = (const float*)d_in[10];
  const float* cr_w1  = (const float*)d_in[11];
  const float* cr_b1  = (const float*)d_in[12];
  const float* cr_w2  = (const float*)d_in[13];
  const float* cr_b2  = (const float*)d_in[14];
  const float* mr_w1  = (const float*)d_in[15];
  const float* mr_b1  = (const float*)d_in[16];
  const float* mr_w2  = (const float*)d_in[17];
  const float* mr_b2  = (const float*)d_in[18];

  const int* src = edge;
  const int* dst = edge + ETOT;

  // workspace layout
  float* deg = (float*)d_ws;                        // 32768
  float* h1  = deg + NTOT;                          // 32768*256
  float* emb = h1 + (size_t)NTOT * DFEAT;           // 32768*256
  float* gcn_w1p = emb + (size_t)NTOT * DFEAT;      // 256*256
  float* gcn_w2p = gcn_w1p + DFEAT * DFEAT;         // 256*256
  float* lr_w1p  = gcn_w2p + DFEAT * DFEAT;         // 256*512
  float* cr_w1p  = lr_w1p + DFEAT * HDIM;
  float* mr_w1p  = cr_w1p + DFEAT * HDIM;
  float* lr_w2p  = mr_w1p + DFEAT * HDIM;           // 512*32
  float* cr_w2p  = lr_w2p + HDIM * 32;
  float* mr_w2p  = cr_w2p + HDIM * 32;
  float* out = (float*)d_out;

  // repack weights into fragment-major layout
  {
    int th_gcn = (DFEAT / 8) * (DFEAT / 16) * 32;   // 16384
    repack_b_kernel<<<(th_gcn + 255) / 256, 256, 0, stream>>>(gcn_w1, gcn_w1p, DFEAT, DFEAT, DFEAT);
    repack_b_kernel<<<(th_gcn + 255) / 256, 256, 0, stream>>>(gcn_w2, gcn_w2p, DFEAT, DFEAT, DFEAT);
    int th_w1 = (DFEAT / 8) * (HDIM / 16) * 32;     // 32768
    repack_b_kernel<<<(th_w1 + 255) / 256, 256, 0, stream>>>(lr_w1, lr_w1p, DFEAT, HDIM, HDIM);
    repack_b_kernel<<<(th_w1 + 255) / 256, 256, 0, stream>>>(cr_w1, cr_w1p, DFEAT, HDIM, HDIM);
    repack_b_kernel<<<(th_w1 + 255) / 256, 256, 0, stream>>>(mr_w1, mr_w1p, DFEAT, HDIM, HDIM);
    int th_w2 = (HDIM / 8) * (32 / 16) * 32;        // 4096
    repack_b_kernel<<<(th_w2 + 255) / 256, 256, 0, stream>>>(lr_w2, lr_w2p, HDIM, 4, 32);
    repack_b_kernel<<<(th_w2 + 255) / 256, 256, 0, stream>>>(cr_w2, cr_w2p, HDIM, 6, 32);
    repack_b_kernel<<<(th_w2 + 255) / 256, 256, 0, stream>>>(mr_w2, mr_w2p, HDIM, 17, 32);
  }

  deg_init_kernel<<<NTOT / 256, 256, 0, stream>>>(deg);
  deg_count_kernel<<<ETOT / 256, 256, 0, stream>>>(dst, deg);

  size_t lds_gcn = (size_t)NPG * DFEAT * sizeof(float);      // 64 KB
  gcn_layer_kernel<true ><<<NGRAPH, 256, lds_gcn, stream>>>(x,  src, dst, deg, gcn_w1p, gcn_b1, h1);
  gcn_layer_kernel<false><<<NGRAPH, 256, lds_gcn, stream>>>(h1, src, dst, deg, gcn_w2p, gcn_b2, emb);

  size_t lds_mlp = (size_t)(64 * DFEAT + 64 * HDIM) * sizeof(float);  // 192 KB
  int nblk = NPAIR / 64;                                              // 4096 blocks
  pair_mlp_kernel<<<nblk, 256, lds_mlp, stream>>>(emb, pairs, lr_w1p, lr_b1, lr_w2p, lr_b2,
                                                  out, 4);
  pair_mlp_kernel<<<nblk, 256, lds_mlp, stream>>>(emb, pairs, cr_w1p, cr_b1, cr_w2p, cr_b2,
                                                  out + (size_t)NPAIR * 4, 6);
  pair_mlp_kernel<<<nblk, 256, lds_mlp, stream>>>(emb, pairs, mr_w1p, mr_b1, mr_w2p, mr_b2,
                                                  out + (size_t)NPAIR * 10, 17);
}